// TGCN_RecurrentGCN_89008902243188
// MI455X (gfx1250) — compile-verified
//
#include <hip/hip_runtime.h>
#include <hip/hip_bf16.h>
#include <math.h>

#define N_NODES 50000
#define NUM_E   1600000
#define F       32
#define NCLS    16
#define NPAD    50048            // 50000 rounded up to multiple of 64 (tile over-read safety)
#define NTILES  3125             // ceil(50000 / 16)

typedef float v2f __attribute__((ext_vector_type(2)));
typedef float v8f __attribute__((ext_vector_type(8)));

__device__ __forceinline__ v8f wmma4(v2f a, v2f b, v8f c) {
    // V_WMMA_F32_16X16X4_F32 : D(16x16,f32) = A(16x4,f32) * B(4x16,f32) + C
    return __builtin_amdgcn_wmma_f32_16x16x4_f32(
        /*neg_a=*/false, a, /*neg_b=*/false, b,
        /*c_mod=*/(short)0, c, /*reuse_a=*/false, /*reuse_b=*/false);
}

// ---------------------------------------------------------------- degree init
__global__ void k_init_deg(float* __restrict__ deg) {
    int i = blockIdx.x * blockDim.x + threadIdx.x;
    if (i < NPAD) deg[i] = (i < N_NODES) ? 1.0f : 0.0f;   // self-loop weight 1
}

// ------------------------------------------------------- degree accumulation
__global__ void k_deg_accum(const int* __restrict__ ei, const float* __restrict__ ew,
                            float* __restrict__ deg) {
    int e = blockIdx.x * blockDim.x + threadIdx.x;
    if (e < NUM_E) atomicAdd(&deg[ei[NUM_E + e]], ew[e]);  // dst row of edge_index
}

// ------------------------------------------------------------------ deg→dinv
__global__ void k_dinv(float* __restrict__ deg) {
    int i = blockIdx.x * blockDim.x + threadIdx.x;
    if (i < NPAD) {
        float d = deg[i];
        deg[i] = (d > 0.0f) ? rsqrtf(d) : 0.0f;
    }
}

// --------------------------- self-loop term; also initializes agg (incl. pad)
__global__ void k_selfloop(const float* __restrict__ x, const float* __restrict__ dinv,
                           float* __restrict__ agg) {
    int t = blockIdx.x * blockDim.x + threadIdx.x;
    if (t >= NPAD * F) return;
    int i = t >> 5;        // node
    int f = t & 31;        // feature
    float dv = dinv[i];
    float v  = (i < N_NODES) ? dv * dv * x[(size_t)i * F + f] : 0.0f;
    agg[(size_t)i * F + f] = v;
}

// -------------------------------------------- edge scatter: wave32 per edge,
// lane = feature -> 128B coalesced gather + contiguous f32 atomic adds
__global__ void k_scatter(const float* __restrict__ x, const int* __restrict__ ei,
                          const float* __restrict__ ew, const float* __restrict__ dinv,
                          float* __restrict__ agg) {
    int tid  = blockIdx.x * blockDim.x + threadIdx.x;
    int e    = tid >> 5;
    int lane = tid & 31;
    if (e >= NUM_E) return;
    int s = ei[e];
    int d = ei[NUM_E + e];
    float nrm = dinv[s] * ew[e] * dinv[d];
    float v   = nrm * x[(size_t)s * F + lane];
    atomicAdd(&agg[(size_t)d * F + lane], v);
}

// ----------------------------------------------------- dense per-node kernel
// one wave32 per 16-node tile: three chained fp32 WMMA matmuls + softmax
__global__ __launch_bounds__(128)
void k_dense(const float* __restrict__ agg,
             const float* __restrict__ Wz, const float* __restrict__ bz,
             const float* __restrict__ Wh, const float* __restrict__ bh,
             const float* __restrict__ Lz, const float* __restrict__ Lzb,
             const float* __restrict__ Lh, const float* __restrict__ Lhb,
             const float* __restrict__ Wo, const float* __restrict__ bo,
             float* __restrict__ out) {
    __shared__ float S[4][16][33];          // per-wave transpose staging, padded
    const int wave = threadIdx.x >> 5;
    const int lane = threadIdx.x & 31;
    const int row  = lane & 15;             // M (A) or N (B/C/D) index
    const bool hi  = lane >= 16;
    const int klo  = hi ? 2 : 0;            // K pair held by this lane-half
    const int moff = hi ? 8 : 0;            // D-layout row offset for upper half
    const int tile = blockIdx.x * 4 + wave; // 16-node tile (pad rows are zero)
    float (*Sw)[33] = S[wave];

    // ---- A operand: 16x32 tile of agg, fp32 A-layout (2 VGPRs per K-chunk)
    v2f a[8];
    const float* ar = agg + (size_t)(tile * 16 + row) * F;
    #pragma unroll
    for (int c = 0; c < 8; ++c) { a[c].x = ar[4*c + klo]; a[c].y = ar[4*c + klo + 1]; }

    // ---- stage 1: Gz = A@Wz + bz ; Gh = A@Wh + bh   (two 16-wide N tiles)
    v8f gz[2], gh[2];
    #pragma unroll
    for (int t = 0; t < 2; ++t) {
        v8f az = {}, ah = {};
        #pragma unroll
        for (int c = 0; c < 8; ++c) {
            int k0 = 4*c + klo;
            v2f b;
            b.x = Wz[k0 * F + t*16 + row]; b.y = Wz[(k0+1) * F + t*16 + row];
            az = wmma4(a[c], b, az);
            b.x = Wh[k0 * F + t*16 + row]; b.y = Wh[(k0+1) * F + t*16 + row];
            ah = wmma4(a[c], b, ah);
        }
        float bzv = bz[t*16 + row], bhv = bh[t*16 + row];
        #pragma unroll
        for (int r = 0; r < 8; ++r) { az[r] += bzv; ah[r] += bhv; }
        gz[t] = az; gh[t] = ah;
    }

    // ---- stage 2a: Z = sigmoid(Gz @ Lz[:32] + Lzb)   (LDS transpose D->A)
    #pragma unroll
    for (int t = 0; t < 2; ++t)
        #pragma unroll
        for (int r = 0; r < 8; ++r) Sw[r + moff][t*16 + row] = gz[t][r];
    __syncthreads();
    v2f a2[8];
    #pragma unroll
    for (int c = 0; c < 8; ++c) { a2[c].x = Sw[row][4*c + klo]; a2[c].y = Sw[row][4*c + klo + 1]; }
    __syncthreads();

    v8f z[2];
    #pragma unroll
    for (int t = 0; t < 2; ++t) {
        v8f acc = {};
        #pragma unroll
        for (int c = 0; c < 8; ++c) {
            int k0 = 4*c + klo;
            v2f b;
            b.x = Lz[k0 * F + t*16 + row]; b.y = Lz[(k0+1) * F + t*16 + row];
            acc = wmma4(a2[c], b, acc);
        }
        float lb = Lzb[t*16 + row];
        #pragma unroll
        for (int r = 0; r < 8; ++r) z[t][r] = 1.0f / (1.0f + __expf(-(acc[r] + lb)));
    }

    // ---- stage 2b: Ht = tanh(Gh @ Lh[:32] + Lhb); Hn = relu((1-Z)*Ht)
    #pragma unroll
    for (int t = 0; t < 2; ++t)
        #pragma unroll
        for (int r = 0; r < 8; ++r) Sw[r + moff][t*16 + row] = gh[t][r];
    __syncthreads();
    v2f a3[8];
    #pragma unroll
    for (int c = 0; c < 8; ++c) { a3[c].x = Sw[row][4*c + klo]; a3[c].y = Sw[row][4*c + klo + 1]; }
    __syncthreads();

    v8f hn[2];
    #pragma unroll
    for (int t = 0; t < 2; ++t) {
        v8f acc = {};
        #pragma unroll
        for (int c = 0; c < 8; ++c) {
            int k0 = 4*c + klo;
            v2f b;
            b.x = Lh[k0 * F + t*16 + row]; b.y = Lh[(k0+1) * F + t*16 + row];
            acc = wmma4(a3[c], b, acc);
        }
        float lb = Lhb[t*16 + row];
        #pragma unroll
        for (int r = 0; r < 8; ++r) {
            float ht = tanhf(acc[r] + lb);
            float v  = (1.0f - z[t][r]) * ht;   // H = (1-Z)*Ht  (H0 == 0)
            hn[t][r] = fmaxf(v, 0.0f);          // relu
        }
    }

    // ---- stage 3: logits = Hn @ Wo + bo  (single 16-class N tile)
    #pragma unroll
    for (int t = 0; t < 2; ++t)
        #pragma unroll
        for (int r = 0; r < 8; ++r) Sw[r + moff][t*16 + row] = hn[t][r];
    __syncthreads();
    v2f a4[8];
    #pragma unroll
    for (int c = 0; c < 8; ++c) { a4[c].x = Sw[row][4*c + klo]; a4[c].y = Sw[row][4*c + klo + 1]; }

    v8f lg = {};
    #pragma unroll
    for (int c = 0; c < 8; ++c) {
        int k0 = 4*c + klo;
        v2f b;
        b.x = Wo[k0 * NCLS + row]; b.y = Wo[(k0+1) * NCLS + row];
        lg = wmma4(a4[c], b, lg);
    }

    // ---- softmax over 16 classes: rows live across 16-lane halves
    float bov = bo[row];
    #pragma unroll
    for (int r = 0; r < 8; ++r) {
        float v  = lg[r] + bov;
        float mx = v;
        mx = fmaxf(mx, __shfl_xor(mx, 1, 32));
        mx = fmaxf(mx, __shfl_xor(mx, 2, 32));
        mx = fmaxf(mx, __shfl_xor(mx, 4, 32));
        mx = fmaxf(mx, __shfl_xor(mx, 8, 32));
        float ex = __expf(v - mx);
        float sm = ex;
        sm += __shfl_xor(sm, 1, 32);
        sm += __shfl_xor(sm, 2, 32);
        sm += __shfl_xor(sm, 4, 32);
        sm += __shfl_xor(sm, 8, 32);
        float p = ex / sm;
        int node = tile * 16 + r + moff;
        if (node < N_NODES) out[(size_t)node * NCLS + row] = p;
    }
}

// ---------------------------------------------------------------------------
extern "C" void kernel_launch(void* const* d_in, const int* in_sizes, int n_in,
                              void* d_out, int out_size, void* d_ws, size_t ws_size,
                              hipStream_t stream) {
    const float* x   = (const float*)d_in[0];
    const int*   ei  = (const int*)  d_in[1];   // [2, E]
    const float* ew  = (const float*)d_in[2];
    const float* Wz  = (const float*)d_in[3];
    const float* bz  = (const float*)d_in[4];
    // d_in[5] (Wr), d_in[6] (br): dead — R is only used in H*R with H==0
    const float* Wh  = (const float*)d_in[7];
    const float* bh  = (const float*)d_in[8];
    const float* Lz  = (const float*)d_in[9];
    const float* Lzb = (const float*)d_in[10];
    // d_in[11] (Lr_w), d_in[12] (Lr_b): dead
    const float* Lh  = (const float*)d_in[13];
    const float* Lhb = (const float*)d_in[14];
    const float* Wo  = (const float*)d_in[15];
    const float* bo  = (const float*)d_in[16];
    float* out = (float*)d_out;

    float* deg = (float*)d_ws;       // NPAD floats: degree, then dinv in place
    float* agg = deg + NPAD;         // NPAD * F floats: aggregated x

    k_init_deg <<<(NPAD + 255) / 256, 256, 0, stream>>>(deg);
    k_deg_accum<<<(NUM_E + 255) / 256, 256, 0, stream>>>(ei, ew, deg);
    k_dinv     <<<(NPAD + 255) / 256, 256, 0, stream>>>(deg);
    k_selfloop <<<(NPAD * F + 255) / 256, 256, 0, stream>>>(x, deg, agg);
    k_scatter  <<<((size_t)NUM_E * 32 + 255) / 256, 256, 0, stream>>>(x, ei, ew, deg, agg);
    k_dense    <<<(NTILES + 3) / 4, 128, 0, stream>>>(agg, Wz, bz, Wh, bh,
                                                      Lz, Lzb, Lh, Lhb, Wo, bo, out);
    (void)in_sizes; (void)n_in; (void)out_size; (void)ws_size;
}